// QEffLlamaSwiftKVModel_21526376088338
// MI455X (gfx1250) — compile-verified
//
#include <hip/hip_runtime.h>
#include <hip/hip_bf16.h>
#include <math.h>

// ---------------- problem constants ----------------
#define B_      8
#define KV_     8192
#define HIDDEN_ 4096            // K and N of both projections
#define NH_     32
#define HD_     128
#define NKV_    8
#define GROUPS_ 4
#define NCHUNK_ 16
#define CHUNK_  (KV_ / NCHUNK_)   // 512
#define KTILE_  32
#define PARTSTRIDE_ 130           // m, l, acc[128]
#define KSPLIT_ 8                 // K-split for the projections
#define KSEG_   (HIDDEN_ / KSPLIT_) // 512

#ifndef __has_builtin
#define __has_builtin(x) 0
#endif

#if __has_builtin(__builtin_amdgcn_global_load_async_to_lds_b128)
#define HAVE_ASYNC_LDS 1
// builtin takes pointers to 16-byte int vectors (gcc vector_size style)
typedef int aiv4 __attribute__((vector_size(16)));
#define ASYNC_G1(p) ((__attribute__((address_space(1))) aiv4*)(p))
#define ASYNC_L3(p) ((__attribute__((address_space(3))) aiv4*)(p))
#else
#define HAVE_ASYNC_LDS 0
#endif

typedef float v2f __attribute__((ext_vector_type(2)));
typedef float v8f __attribute__((ext_vector_type(8)));

__device__ __forceinline__ void wait_async_zero() {
#if __has_builtin(__builtin_amdgcn_s_wait_asynccnt)
    __builtin_amdgcn_s_wait_asynccnt(0);
#elif defined(__AMDGCN__)
    asm volatile("s_wait_asynccnt 0x0" ::: "memory");
#endif
}

// =====================================================================
// Kernel 1: K-split partial GEMM:  Ypart[ks][0..B_)[N] =
//   X(B_ x 4096, padded to M=16) @ W(4096 x 4096), K-segment ks.
// One wave per (16-column tile, K-segment). fp32 WMMA 16x16x4.
// A layout (32-bit A 16x4): lanes 0-15 M=0..15; VGPR0/1 = K 0/1 (K 2/3 upper half-wave)
// B layout (32-bit B 4x16): lanes 0-15 N=0..15; VGPR0/1 = K 0/1 (K 2/3 upper half-wave)
// C/D: VGPR v = row (lane<16 ? v : v+8), col = lane&15.
// Rows are compile-time B_=8: lanes 0-15 own C rows 0-7 (all stored),
// lanes 16-31 own C rows 8-15 (padding, discarded) -> uniform store mask.
// =====================================================================
__global__ __launch_bounds__(32) void gemm_rows_wmma(
    const float* __restrict__ X, const float* __restrict__ W,
    float* __restrict__ Ypart)
{
    const int lane = threadIdx.x & 31;
    const int m    = lane & 15;
    const int kO   = (lane >> 4) << 1;                 // 0 or 2
    const int n    = blockIdx.x * 16 + m;
    const int k0   = blockIdx.y * KSEG_;
    // clamp instead of predicate: padding rows only pollute C rows we discard
    const int arow = (m < B_) ? m : (B_ - 1);

    const float* __restrict__ xp = X + (size_t)arow * HIDDEN_ + k0 + kO;
    const float* __restrict__ wp = W + (size_t)(k0 + kO) * HIDDEN_ + n;

    v8f c = {0.f,0.f,0.f,0.f,0.f,0.f,0.f,0.f};

#if __has_builtin(__builtin_amdgcn_wmma_f32_16x16x4_f32)
    #pragma unroll 8
    for (int kk = 0; kk < KSEG_; kk += 4) {
        if ((kk & 31) == 0)   // one prefetch per 8 iterations, ~16 iters ahead
            __builtin_prefetch(wp + (size_t)(kk + 64) * HIDDEN_, 0, 3);
        const v2f a = *(const v2f*)(xp + kk);
        v2f b;
        b.x = wp[(size_t)kk * HIDDEN_];
        b.y = wp[(size_t)kk * HIDDEN_ + HIDDEN_];
        c = __builtin_amdgcn_wmma_f32_16x16x4_f32(
                /*neg_a=*/false, a, /*neg_b=*/false, b,
                /*c_mod=*/(short)0, c, /*reuse_a=*/false, /*reuse_b=*/false);
    }
#else
    // Scalar fallback with identical lane-ownership of C (host pass safety)
    for (int kk = 0; kk < KSEG_; ++kk) {
        const float w = W[(size_t)(k0 + kk) * HIDDEN_ + n];
        #pragma unroll
        for (int v = 0; v < 8; ++v) {
            const int row = ((lane >> 4) << 3) + v;
            const float x = (row < B_) ? X[(size_t)row * HIDDEN_ + k0 + kk] : 0.f;
            c[v] += x * w;
        }
    }
#endif

    // lanes 0-15 hold rows 0-7 (valid); lanes 16-31 hold rows 8-15 (padding)
    if (lane < 16) {
        float* __restrict__ dst = Ypart + (size_t)blockIdx.y * B_ * HIDDEN_ + n;
        #pragma unroll
        for (int v = 0; v < 8; ++v)
            dst[(size_t)v * HIDDEN_] = c[v];
    }
}

// Deterministic K-split reduction: out[i] = sum_ks part[ks][i]
__global__ __launch_bounds__(256) void reduce_ksplit(
    const float* __restrict__ part, float* __restrict__ out, int n)
{
    const int i = blockIdx.x * 256 + threadIdx.x;
    if (i < n) {
        float s = 0.f;
        #pragma unroll
        for (int ks = 0; ks < KSPLIT_; ++ks) s += part[(size_t)ks * n + i];
        out[i] = s;
    }
}

// =====================================================================
// Kernel 2: flash-decode partials.
// grid = B*NKV*NCHUNK blocks of 256 threads. Each block: one kv-head group
// (4 query heads) x one 512-key chunk. Emits (m, l, acc[128]) per head.
// =====================================================================
__global__ __launch_bounds__(256) void attn_partial(
    const float* __restrict__ qbuf,        // [B][NH*HD] (pre-RoPE)
    const int*   __restrict__ position_ids,// [B]
    const float* __restrict__ key_cache,   // [B][NKV][KV][HD]
    const float* __restrict__ value_cache, // [B][NKV][KV][HD]
    float* __restrict__ part)              // [B*NH][NCHUNK][130]
{
    __shared__ float qs[4][HD_];             // roped + scaled queries
    __shared__ float kbuf[KTILE_][HD_ + 4];  // staged K tile (padded rows)
    __shared__ float sbuf[4][CHUNK_];        // scores -> probabilities
    __shared__ float red[256];
    __shared__ float vred[8 * HD_];
    __shared__ float ml[4][2];

    const int bid   = blockIdx.x;
    const int chunk = bid % NCHUNK_;
    const int kvh   = (bid / NCHUNK_) % NKV_;
    const int b     = bid / (NCHUNK_ * NKV_);
    const int tid   = threadIdx.x;
    const int pos   = position_ids[b];
    const int j0    = chunk * CHUNK_;

    // ---- fully-masked chunk: emit empty partial and exit ----
    if (j0 > pos) {
        for (int i = tid; i < 4 * PARTSTRIDE_; i += 256) {
            const int hh = i / PARTSTRIDE_, r = i % PARTSTRIDE_;
            float* dst = part + ((size_t)(b * NH_ + kvh * GROUPS_ + hh) * NCHUNK_ + chunk)
                              * PARTSTRIDE_;
            dst[r] = (r == 0) ? -1e30f : 0.0f;
        }
        return;
    }

    // ---- load q for the 4 grouped heads, apply RoPE and 1/sqrt(HD) ----
    const float scale = 0.08838834764831845f; // 1/sqrt(128)
    for (int i = tid; i < 4 * HD_; i += 256) {
        const int hh = i / HD_, d = i % HD_;
        const int h  = kvh * GROUPS_ + hh;
        const float qv = qbuf[b * (NH_ * HD_) + h * HD_ + d];
        const int  di  = (d < 64) ? d : d - 64;
        const float inv_freq = powf(10000.0f, -(float)(2 * di) / 128.0f);
        const float ang = (float)pos * inv_freq;
        const float cv = cosf(ang), sv = sinf(ang);
        const float rot = (d < 64)
            ? -qbuf[b * (NH_ * HD_) + h * HD_ + d + 64]
            :  qbuf[b * (NH_ * HD_) + h * HD_ + d - 64];
        qs[hh][d] = (qv * cv + rot * sv) * scale;
    }
    __syncthreads();

    // ---- score pass: async-stage K tiles to LDS, 8 lanes per key row ----
    const float* __restrict__ Kbase =
        key_cache + (size_t)(b * NKV_ + kvh) * KV_ * HD_;
    const int wave = tid >> 5, lane = tid & 31;
    const int g = lane >> 3, r = lane & 7;

    for (int t0 = 0; t0 < CHUNK_; t0 += KTILE_) {
        // 32 rows x 128 floats; each thread moves 4x b128 (coalesced, ASYNCcnt)
        for (int li = tid; li < KTILE_ * 32; li += 256) {
            const int row = li >> 5, quad = li & 31;
            const float* gsrc = Kbase + (size_t)(j0 + t0 + row) * HD_ + quad * 4;
            float* ldst = &kbuf[row][quad * 4];
#if HAVE_ASYNC_LDS
            __builtin_amdgcn_global_load_async_to_lds_b128(
                ASYNC_G1(gsrc), ASYNC_L3(ldst), 0, 0);
#else
            *(float4*)ldst = *(const float4*)gsrc;
#endif
        }
#if HAVE_ASYNC_LDS
        wait_async_zero();
#endif
        __syncthreads();

        const int row = wave * 4 + g;       // 8 waves * 4 groups = 32 rows
        const int j   = t0 + row;           // key index within chunk
        float a0 = 0.f, a1 = 0.f, a2 = 0.f, a3 = 0.f;
        #pragma unroll
        for (int q4 = 0; q4 < 4; ++q4) {
            const int d = r * 16 + q4 * 4;
            const float4 kv = *(const float4*)&kbuf[row][d];
            const float4 q0 = *(const float4*)&qs[0][d];
            const float4 q1 = *(const float4*)&qs[1][d];
            const float4 q2 = *(const float4*)&qs[2][d];
            const float4 q3 = *(const float4*)&qs[3][d];
            a0 += kv.x*q0.x + kv.y*q0.y + kv.z*q0.z + kv.w*q0.w;
            a1 += kv.x*q1.x + kv.y*q1.y + kv.z*q1.z + kv.w*q1.w;
            a2 += kv.x*q2.x + kv.y*q2.y + kv.z*q2.z + kv.w*q2.w;
            a3 += kv.x*q3.x + kv.y*q3.y + kv.z*q3.z + kv.w*q3.w;
        }
        #pragma unroll
        for (int off = 4; off >= 1; off >>= 1) {
            a0 += __shfl_xor(a0, off);
            a1 += __shfl_xor(a1, off);
            a2 += __shfl_xor(a2, off);
            a3 += __shfl_xor(a3, off);
        }
        if (r == 0) {
            const bool ok = (j0 + j) <= pos;
            sbuf[0][j] = ok ? a0 : -1e9f;
            sbuf[1][j] = ok ? a1 : -1e9f;
            sbuf[2][j] = ok ? a2 : -1e9f;
            sbuf[3][j] = ok ? a3 : -1e9f;
        }
        __syncthreads();
    }

    // ---- chunk softmax: m and l per head (64 threads per head) ----
    const int hh64 = tid >> 6, li = tid & 63;
    float mx = -1e30f;
    for (int j = li; j < CHUNK_; j += 64) mx = fmaxf(mx, sbuf[hh64][j]);
    red[tid] = mx; __syncthreads();
    for (int off = 32; off >= 1; off >>= 1) {
        if (li < off) red[tid] = fmaxf(red[tid], red[tid + off]);
        __syncthreads();
    }
    const float m_h = red[hh64 << 6];
    __syncthreads();

    float sm = 0.f;
    for (int j = li; j < CHUNK_; j += 64) {
        const float p = __expf(sbuf[hh64][j] - m_h);
        sbuf[hh64][j] = p;
        sm += p;
    }
    red[tid] = sm; __syncthreads();
    for (int off = 32; off >= 1; off >>= 1) {
        if (li < off) red[tid] += red[tid + off];
        __syncthreads();
    }
    if (li == 0) { ml[hh64][0] = m_h; ml[hh64][1] = red[tid]; }
    __syncthreads();

    // ---- V pass: acc[d] = sum_j p_j * V[j][d], coalesced float4 rows ----
    const float* __restrict__ Vbase =
        value_cache + (size_t)(b * NKV_ + kvh) * KV_ * HD_;
    const int dq = tid & 31;   // float4 slot along HD
    const int jg = tid >> 5;   // 8-way split over keys

    for (int hh = 0; hh < 4; ++hh) {
        float4 acc = make_float4(0.f, 0.f, 0.f, 0.f);
        for (int j = jg; j < CHUNK_; j += 8) {
            const float p = sbuf[hh][j];
            const float4 v4 = *(const float4*)(Vbase +
                (size_t)(j0 + j) * HD_ + dq * 4);
            acc.x += p * v4.x; acc.y += p * v4.y;
            acc.z += p * v4.z; acc.w += p * v4.w;
        }
        *(float4*)&vred[jg * HD_ + dq * 4] = acc;
        __syncthreads();
        #pragma unroll
        for (int off = 4; off >= 1; off >>= 1) {
            if (jg < off) {
                float4 x = *(float4*)&vred[jg * HD_ + dq * 4];
                const float4 y = *(const float4*)&vred[(jg + off) * HD_ + dq * 4];
                x.x += y.x; x.y += y.y; x.z += y.z; x.w += y.w;
                *(float4*)&vred[jg * HD_ + dq * 4] = x;
            }
            __syncthreads();
        }
        if (jg == 0) {
            float* dst = part + ((size_t)(b * NH_ + kvh * GROUPS_ + hh) * NCHUNK_ + chunk)
                              * PARTSTRIDE_;
            const float4 x = *(const float4*)&vred[dq * 4];
            dst[2 + dq * 4 + 0] = x.x;
            dst[2 + dq * 4 + 1] = x.y;
            dst[2 + dq * 4 + 2] = x.z;
            dst[2 + dq * 4 + 3] = x.w;
            if (dq == 0) { dst[0] = ml[hh][0]; dst[1] = ml[hh][1]; }
        }
        __syncthreads();
    }
}

// =====================================================================
// Kernel 3: merge chunk partials. grid = B*NH blocks of HD threads.
// =====================================================================
__global__ __launch_bounds__(128) void attn_combine(
    const float* __restrict__ part, float* __restrict__ attn_out)
{
    const int bh = blockIdx.x;          // b*NH + h
    const int d  = threadIdx.x;         // 0..127
    const float* p = part + (size_t)bh * NCHUNK_ * PARTSTRIDE_;

    float M = -1e30f;
    #pragma unroll
    for (int c = 0; c < NCHUNK_; ++c) M = fmaxf(M, p[c * PARTSTRIDE_]);
    float L = 0.f, acc = 0.f;
    #pragma unroll
    for (int c = 0; c < NCHUNK_; ++c) {
        const float w = __expf(p[c * PARTSTRIDE_] - M);
        L   += w * p[c * PARTSTRIDE_ + 1];
        acc += w * p[c * PARTSTRIDE_ + 2 + d];
    }
    attn_out[bh * HD_ + d] = acc / L;   // [B][NH*HD] row-major == bh*HD+d
}

// =====================================================================
// launcher
// =====================================================================
extern "C" void kernel_launch(void* const* d_in, const int* in_sizes, int n_in,
                              void* d_out, int out_size, void* d_ws, size_t ws_size,
                              hipStream_t stream) {
    const float* hidden = (const float*)d_in[0];   // [8,1,4096]
    const int*   pid    = (const int*)  d_in[1];   // [8,1] (int32)
    const float* kcache = (const float*)d_in[2];   // [8,8,8192,128]
    const float* vcache = (const float*)d_in[3];   // [8,8,8192,128]
    // d_in[4] attention_mask: unused (recomputed exactly from position_ids)
    const float* w_q    = (const float*)d_in[5];   // [4096,4096]
    const float* w_o    = (const float*)d_in[6];   // [4096,4096]
    float* out = (float*)d_out;                    // [8,1,4096]

    const int NOUT = B_ * HIDDEN_;                 // 32768
    float* ws       = (float*)d_ws;
    float* qbuf     = ws;                          // 32768
    float* attn_out = qbuf + NOUT;                 // 32768
    float* partbuf  = attn_out + NOUT;             // 8*32*16*130  (~2.1MB)
    float* gpart    = partbuf + (size_t)B_ * NH_ * NCHUNK_ * PARTSTRIDE_; // 8*32768 (1MB)

    // 1) q projection (pre-RoPE): [8,4096] = hidden @ w_q (K split 8 ways)
    gemm_rows_wmma<<<dim3(HIDDEN_ / 16, KSPLIT_), 32, 0, stream>>>(hidden, w_q, gpart);
    reduce_ksplit<<<(NOUT + 255) / 256, 256, 0, stream>>>(gpart, qbuf, NOUT);
    // 2) flash-decode partials over split KV
    attn_partial<<<B_ * NKV_ * NCHUNK_, 256, 0, stream>>>(qbuf, pid, kcache,
                                                          vcache, partbuf);
    // 3) merge partials
    attn_combine<<<B_ * NH_, 128, 0, stream>>>(partbuf, attn_out);
    // 4) output projection: out = attn_out @ w_o (K split 8 ways)
    gemm_rows_wmma<<<dim3(HIDDEN_ / 16, KSPLIT_), 32, 0, stream>>>(attn_out, w_o, gpart);
    reduce_ksplit<<<(NOUT + 255) / 256, 256, 0, stream>>>(gpart, out, NOUT);
}